// stochastic_opt_model_23287312679684
// MI455X (gfx1250) — compile-verified
//
#include <hip/hip_runtime.h>
#include <math.h>

#define NG 24   // generators (QP dimension)
#define NC 23   // ramp constraints
#define NEWTON 11   // 10 detached Newton steps + 1 final (forward values identical)
#define IPM_ITERS 30
#define BLOCK 64

#if __has_builtin(__builtin_amdgcn_global_load_async_to_lds_b128)
#define HAVE_ASYNC_LDS 1
typedef int v4i __attribute__((ext_vector_type(4)));
typedef __attribute__((address_space(1))) v4i* gv4p;   // global int4*
typedef __attribute__((address_space(3))) v4i* lv4p;   // LDS int4*
#else
#define HAVE_ASYNC_LDS 0
#endif

__device__ __forceinline__ void wait_asynccnt0() {
#if __has_builtin(__builtin_amdgcn_s_wait_asynccnt)
  __builtin_amdgcn_s_wait_asynccnt(0);
#else
  asm volatile("s_wait_asynccnt 0x0" ::: "memory");
#endif
}

__device__ __forceinline__ float frcp(float x) {
#if __has_builtin(__builtin_amdgcn_rcpf)
  return __builtin_amdgcn_rcpf(x);   // v_rcp_f32
#else
  return 1.0f / x;
#endif
}

__global__ __launch_bounds__(BLOCK, 1)
void stochastic_opt_ipm_kernel(const float* __restrict__ x_in,
                               const float* __restrict__ mu_in,
                               const float* __restrict__ sg_in,
                               float* __restrict__ out, int B) {
  const int b = blockIdx.x * blockDim.x + threadIdx.x;
  if (b >= B) return;

  const float C_RAMP = 0.4f;
  const float GSUM = 50.5f;              // GAMMA_UNDER + GAMMA_OVER
  const float GU = 50.0f;
  const float INV_SQRT2 = 0.7071067811865476f;
  const float INV_SQRT_2PI = 0.3989422804014327f;

  float x[NG], mu[NG], sg[NG];

#if HAVE_ASYNC_LDS
  // Stage each lane's three 96B rows through LDS with the CDNA5 async path
  // (GLOBAL_LOAD_ASYNC_TO_LDS_B128, tracked by ASYNCcnt).
  __shared__ __align__(16) float tX[BLOCK * NG];
  __shared__ __align__(16) float tM[BLOCK * NG];
  __shared__ __align__(16) float tS[BLOCK * NG];
  {
    const int tid = threadIdx.x;
    const float* gx = x_in + (size_t)b * NG;
    const float* gm = mu_in + (size_t)b * NG;
    const float* gs = sg_in + (size_t)b * NG;
    float* lx = &tX[tid * NG];
    float* lm = &tM[tid * NG];
    float* ls = &tS[tid * NG];
#define ASYNC_B128(G, L, Q)                                                    \
    __builtin_amdgcn_global_load_async_to_lds_b128(                            \
        (gv4p)((G) + 4 * (Q)), (lv4p)((L) + 4 * (Q)), 0, 0)
    ASYNC_B128(gx, lx, 0); ASYNC_B128(gx, lx, 1); ASYNC_B128(gx, lx, 2);
    ASYNC_B128(gx, lx, 3); ASYNC_B128(gx, lx, 4); ASYNC_B128(gx, lx, 5);
    ASYNC_B128(gm, lm, 0); ASYNC_B128(gm, lm, 1); ASYNC_B128(gm, lm, 2);
    ASYNC_B128(gm, lm, 3); ASYNC_B128(gm, lm, 4); ASYNC_B128(gm, lm, 5);
    ASYNC_B128(gs, ls, 0); ASYNC_B128(gs, ls, 1); ASYNC_B128(gs, ls, 2);
    ASYNC_B128(gs, ls, 3); ASYNC_B128(gs, ls, 4); ASYNC_B128(gs, ls, 5);
#undef ASYNC_B128
    wait_asynccnt0();
    const float4* xv = reinterpret_cast<const float4*>(lx);
    const float4* mv = reinterpret_cast<const float4*>(lm);
    const float4* sv = reinterpret_cast<const float4*>(ls);
#pragma unroll
    for (int q = 0; q < NG / 4; ++q) {
      float4 a = xv[q]; x[4*q] = a.x; x[4*q+1] = a.y; x[4*q+2] = a.z; x[4*q+3] = a.w;
      float4 c = mv[q]; mu[4*q] = c.x; mu[4*q+1] = c.y; mu[4*q+2] = c.z; mu[4*q+3] = c.w;
      float4 d = sv[q]; sg[4*q] = d.x; sg[4*q+1] = d.y; sg[4*q+2] = d.z; sg[4*q+3] = d.w;
    }
  }
#else
  // Fallback: direct b128 loads (each lane's 96B row is contiguous; a wave's
  // access is a dense 3KB span, full cacheline utilization).
  __builtin_prefetch(mu_in + (size_t)b * NG, 0, 2);   // global_prefetch_b8
  __builtin_prefetch(sg_in + (size_t)b * NG, 0, 2);
  {
    const float4* xv = reinterpret_cast<const float4*>(x_in + (size_t)b * NG);
    const float4* mv = reinterpret_cast<const float4*>(mu_in + (size_t)b * NG);
    const float4* sv = reinterpret_cast<const float4*>(sg_in + (size_t)b * NG);
#pragma unroll
    for (int q = 0; q < NG / 4; ++q) {
      float4 a = xv[q]; x[4*q] = a.x; x[4*q+1] = a.y; x[4*q+2] = a.z; x[4*q+3] = a.w;
      float4 c = mv[q]; mu[4*q] = c.x; mu[4*q+1] = c.y; mu[4*q+2] = c.z; mu[4*q+3] = c.w;
      float4 d = sv[q]; sg[4*q] = d.x; sg[4*q+1] = d.y; sg[4*q+2] = d.z; sg[4*q+3] = d.w;
    }
  }
#endif

  for (int t = 0; t < NEWTON; ++t) {
    // ---- build QP: qd = pdf*(GU+GO)+1, p = cdf*(GU+GO)-GU + x-mu ----
    float qd[NG], p[NG];
#pragma unroll
    for (int j = 0; j < NG; ++j) {
      float rs = frcp(sg[j]);
      float zs = (x[j] - mu[j]) * rs;
      float cdf = 0.5f * (1.0f + erff(zs * INV_SQRT2));
      float pdf = __expf(-0.5f * zs * zs) * INV_SQRT_2PI * rs;
      qd[j] = pdf * GSUM + 1.0f;
      p[j]  = cdf * GSUM - GU + x[j] - mu[j];
    }

    // ---- primal-dual IPM, fixed 30 iterations ----
    float z[NG], s[NC], lam[NC];
#pragma unroll
    for (int j = 0; j < NG; ++j) z[j] = 0.0f;
#pragma unroll
    for (int i = 0; i < NC; ++i) { s[i] = C_RAMP; lam[i] = 1.0f; }

    for (int it = 0; it < IPM_ITERS; ++it) {
      // duality gap / centering
      float sl = 0.0f;
#pragma unroll
      for (int i = 0; i < NC; ++i) sl += s[i] * lam[i];
      const float cmu = 0.1f * sl * (1.0f / (float)NC);   // 0.1 * mu

      // D = lam/s ; rcs = r_comp/s = lam - 0.1*mu/s ; rp = Gz + s - h
      float D[NC], rp[NC], rcs[NC];
#pragma unroll
      for (int i = 0; i < NC; ++i) {
        float rs = frcp(s[i]);
        D[i]   = lam[i] * rs;
        rcs[i] = lam[i] - cmu * rs;
        rp[i]  = (z[i + 1] - z[i]) + s[i] - C_RAMP;
      }

      // Thomas forward sweep on tridiagonal M = diag(qd) + G^T D G,
      // rhs_j = -(qd*z + p + (lam[j-1]-lam[j]) + (w[j-1]-w[j])),
      // w_i = D_i*rp_i - rcs_i, fused via scalar carries.
      float dp_[NG], cp[NC];
      {
        float lamprev = 0.0f, wprev = 0.0f, Dprev = 0.0f;
        float prevc = 0.0f, prevd = 0.0f;
#pragma unroll
        for (int j = 0; j < NG; ++j) {
          float l0 = (j < NC) ? lam[j] : 0.0f;
          float w0 = (j < NC) ? (D[j] * rp[j] - rcs[j]) : 0.0f;
          float Dj = (j < NC) ? D[j] : 0.0f;
          float rhsj = -(qd[j] * z[j] + p[j] + (lamprev - l0) + (wprev - w0));
          float md = qd[j] + Dprev + Dj;        // main diagonal
          float aj = -Dprev;                    // sub-diagonal  (-D[j-1])
          float rd = frcp(md - aj * prevc);
          float cj = -Dj * rd;                  // modified upper coeff
          float dj = (rhsj - aj * prevd) * rd;
          if (j < NC) cp[j] = cj;
          dp_[j] = dj;
          prevc = cj; prevd = dj;
          lamprev = l0; wprev = w0; Dprev = Dj;
        }
        // back substitution: dp_ becomes dz
#pragma unroll
        for (int j = NG - 2; j >= 0; --j) dp_[j] = dp_[j] - cp[j] * dp_[j + 1];
      }

      // fraction-to-boundary step length (dlam/ds computed on the fly)
      float a_l = 1e10f, a_s = 1e10f;
#pragma unroll
      for (int i = 0; i < NC; ++i) {
        float dl  = D[i] * ((dp_[i + 1] - dp_[i]) + rp[i]) - rcs[i];
        float ds_ = -s[i] * (rcs[i] + dl) * frcp(lam[i]);
        if (dl  < 0.0f) a_l = fminf(a_l, -lam[i] * frcp(dl));
        if (ds_ < 0.0f) a_s = fminf(a_s, -s[i] * frcp(ds_));
      }
      const float alpha = fminf(1.0f, 0.99f * fminf(a_l, a_s));

      // apply step (recompute dlam/ds; cheaper than keeping them live)
#pragma unroll
      for (int j = 0; j < NG; ++j) z[j] += alpha * dp_[j];
#pragma unroll
      for (int i = 0; i < NC; ++i) {
        float dl  = D[i] * ((dp_[i + 1] - dp_[i]) + rp[i]) - rcs[i];
        float ds_ = -s[i] * (rcs[i] + dl) * frcp(lam[i]);
        s[i]   += alpha * ds_;
        lam[i] += alpha * dl;
      }
    }

    // Newton update x += z  (11th update IS the output value)
#pragma unroll
    for (int j = 0; j < NG; ++j) x[j] += z[j];
  }

  {
    float4* ov = reinterpret_cast<float4*>(out + (size_t)b * NG);
#pragma unroll
    for (int q = 0; q < NG / 4; ++q) {
      float4 a;
      a.x = x[4*q]; a.y = x[4*q+1]; a.z = x[4*q+2]; a.w = x[4*q+3];
      ov[q] = a;
    }
  }
}

extern "C" void kernel_launch(void* const* d_in, const int* in_sizes, int n_in,
                              void* d_out, int out_size, void* d_ws, size_t ws_size,
                              hipStream_t stream) {
  (void)n_in; (void)d_ws; (void)ws_size; (void)out_size;
  const float* x  = (const float*)d_in[0];
  const float* mu = (const float*)d_in[1];
  const float* sg = (const float*)d_in[2];
  float* out = (float*)d_out;
  const int B = in_sizes[0] / NG;          // 16384 samples
  const int blocks = (B + BLOCK - 1) / BLOCK;
  stochastic_opt_ipm_kernel<<<blocks, BLOCK, 0, stream>>>(x, mu, sg, out, B);
}